// RiskGNN_86122684219831
// MI455X (gfx1250) — compile-verified
//
#include <hip/hip_runtime.h>
#include <math.h>

// ---------------------------------------------------------------------------
// GATv2 x3 + classifier for MI455X (gfx1250, wave32).
// Dense node-feature GEMMs use V_WMMA_F32_16X16X4_F32 (exact fp32 WMMA),
// branchless loads (clamped indices, store-side guard only).
// Edge phase (gather / segment-softmax / scatter) is HBM/L2-bandwidth bound;
// vectorized float4 gathers, templated head/channel counts.
// ---------------------------------------------------------------------------

typedef __attribute__((ext_vector_type(2))) float v2f;
typedef __attribute__((ext_vector_type(8))) float v8f;

#define NEG_SLOPE 0.2f

__device__ __forceinline__ int imin(int a, int b) { return a < b ? a : b; }

__device__ __forceinline__ int ordf(float f) {
  int i = __float_as_int(f);
  return (i >= 0) ? i : (i ^ 0x7FFFFFFF);   // monotonic float -> int
}
__device__ __forceinline__ float deordf(int i) {
  int j = (i >= 0) ? i : (i ^ 0x7FFFFFFF);
  return __int_as_float(j);
}

// ---------------------------------------------------------------- utilities
__global__ void fill_u32_kernel(unsigned* p, unsigned v, long long n) {
  long long i = (long long)blockIdx.x * blockDim.x + threadIdx.x;
  long long stride = (long long)gridDim.x * blockDim.x;
  for (; i < n; i += stride) p[i] = v;
}

// ------------------------------------------------------- self-loop edge attr
__global__ void sl_accum_kernel(const int* __restrict__ dst,
                                const float* __restrict__ ea,
                                float* __restrict__ asum,
                                float* __restrict__ cnt, int E) {
  int i = blockIdx.x * blockDim.x + threadIdx.x;
  if (i >= E) return;
  int d = dst[i];
  atomicAdd(&asum[d], ea[i]);
  atomicAdd(&cnt[d], 1.0f);
}

__global__ void sl_final_kernel(const float* __restrict__ asum,
                                const float* __restrict__ cnt,
                                float* __restrict__ loopA, int N) {
  int i = blockIdx.x * blockDim.x + threadIdx.x;
  if (i >= N) return;
  loopA[i] = asum[i] / fmaxf(cnt[i], 1.0f);
}

// ------------------------------------------------------------- fp32 WMMA GEMM
// Y[Mrows x Ncols] = X[Mrows x K] @ W[K x Ncols], K % 4 == 0 (compile-time).
// One wave per 16x16 output tile, K stepped by 4 via V_WMMA_F32_16X16X4_F32.
// Branchless: out-of-range rows/cols are clamped (a bad A-row only corrupts
// that output row; a bad B-col only corrupts that output col) and masked at
// the store. A-frag: lanes 0-15 K={k,k+1}, lanes 16-31 K={k+2,k+3}.
template <int K>
__global__ void wmma_gemm_kernel(const float* __restrict__ X,
                                 const float* __restrict__ W,
                                 float* __restrict__ Y,
                                 int Mrows, int Ncols, int tilesN) {
  int tilesM = (Mrows + 15) >> 4;
  int t = blockIdx.x * blockDim.y + threadIdx.y;
  if (t >= tilesM * tilesN) return;
  int tm = t / tilesN, tn = t % tilesN;
  int lane = threadIdx.x;
  int half = lane >> 4;       // 0: lanes 0-15, 1: lanes 16-31
  int l    = lane & 15;
  int row0 = tm * 16, col0 = tn * 16;
  int ar  = imin(row0 + l, Mrows - 1);   // clamped (store-guarded)
  int bc0 = col0 + l;
  int bc  = imin(bc0, Ncols - 1);        // clamped (store-guarded)
  const float* xrow = X + (long long)ar * K + 2 * half;
  const float* wcol = W + bc + (long long)2 * half * Ncols;
  v8f acc = {};
#pragma unroll 8
  for (int k = 0; k < K; k += 4) {
    v2f a = *(const v2f*)(xrow + k);     // contiguous -> global_load_b64
    v2f b;
    b.x = wcol[(long long)k * Ncols];
    b.y = wcol[(long long)(k + 1) * Ncols];
    acc = __builtin_amdgcn_wmma_f32_16x16x4_f32(
        /*neg_a=*/false, a, /*neg_b=*/false, b,
        /*c_mod=*/(short)0, acc, /*reuse_a=*/false, /*reuse_b=*/false);
  }
  if (bc0 < Ncols) {
#pragma unroll
    for (int r = 0; r < 8; ++r) {        // C/D: VGPR r -> M=r (+8 upper half)
      int row = row0 + half * 8 + r;
      if (row < Mrows) Y[(long long)row * Ncols + bc0] = acc[r];
    }
  }
}

// ------------------------------------------------------------ edge kernels
// Edges 0..E-1 from edge_index; edges E..E+N-1 are self loops (attr = loopA).

// s[e,h] = sum_c leaky_relu(xl[s][h,c] + xr[d][h,c] + ea*We[h,c]) * att[h,c]
template <int H, int C>
__global__ void edge_score_kernel(const int* __restrict__ src,
                                  const int* __restrict__ dst,
                                  const float* __restrict__ ea,
                                  const float* __restrict__ loopA,
                                  const float* __restrict__ XL,
                                  const float* __restrict__ XR,
                                  const float* __restrict__ We,
                                  const float* __restrict__ att,
                                  float* __restrict__ S,
                                  int* __restrict__ Mx,
                                  int E, int N) {
  constexpr int F = H * C;
  long long idx = (long long)blockIdx.x * blockDim.x + threadIdx.x;
  long long total = (long long)(E + N) * H;
  if (idx >= total) return;
  int e = (int)(idx / H), h = (int)(idx % H);
  int s, d; float a;
  if (e < E) { s = src[e]; d = dst[e]; a = ea[e]; }
  else       { s = d = e - E;          a = loopA[e - E]; }
  const float4* xl4 = (const float4*)(XL + (long long)s * F + h * C);
  const float4* xr4 = (const float4*)(XR + (long long)d * F + h * C);
  const float4* we4 = (const float4*)(We + h * C);
  const float4* at4 = (const float4*)(att + h * C);
  float acc = 0.0f;
#pragma unroll
  for (int q = 0; q < C / 4; ++q) {
    float4 vl = xl4[q], vr = xr4[q], vw = we4[q], va = at4[q];
    float m0 = vl.x + vr.x + a * vw.x; m0 = (m0 > 0.f) ? m0 : NEG_SLOPE * m0;
    float m1 = vl.y + vr.y + a * vw.y; m1 = (m1 > 0.f) ? m1 : NEG_SLOPE * m1;
    float m2 = vl.z + vr.z + a * vw.z; m2 = (m2 > 0.f) ? m2 : NEG_SLOPE * m2;
    float m3 = vl.w + vr.w + a * vw.w; m3 = (m3 > 0.f) ? m3 : NEG_SLOPE * m3;
    acc += m0 * va.x + m1 * va.y + m2 * va.z + m3 * va.w;
  }
  S[idx] = acc;
  atomicMax(&Mx[d * H + h], ordf(acc));
}

// ex = exp(s - m[dst]);  den[dst] += ex
template <int H>
__global__ void edge_exp_kernel(const int* __restrict__ dst,
                                float* __restrict__ S,
                                const int* __restrict__ Mx,
                                float* __restrict__ Den,
                                int E, int N) {
  long long idx = (long long)blockIdx.x * blockDim.x + threadIdx.x;
  long long total = (long long)(E + N) * H;
  if (idx >= total) return;
  int e = (int)(idx / H), h = (int)(idx % H);
  int d = (e < E) ? dst[e] : (e - E);
  float m = deordf(Mx[d * H + h]);
  if (!isfinite(m)) m = 0.0f;
  float ex = expf(S[idx] - m);
  S[idx] = ex;
  atomicAdd(&Den[d * H + h], ex);
}

// out[dst] += (ex/den[dst]) * xl[src]   (one thread per edge-feature)
template <int H, int C>
__global__ void edge_aggr_kernel(const int* __restrict__ src,
                                 const int* __restrict__ dst,
                                 const float* __restrict__ S,
                                 const float* __restrict__ Den,
                                 const float* __restrict__ XL,
                                 float* __restrict__ Out,
                                 int E, int N) {
  constexpr int F = H * C;                 // power of two
  long long idx = (long long)blockIdx.x * blockDim.x + threadIdx.x;
  long long total = (long long)(E + N) * F;
  if (idx >= total) return;
  int e = (int)(idx / F), f = (int)(idx % F);
  int h = f / C;
  int s, d;
  if (e < E) { s = src[e]; d = dst[e]; }
  else       { s = d = e - E; }
  float alpha = S[(long long)e * H + h] / (Den[d * H + h] + 1e-16f);
  atomicAdd(&Out[(long long)d * F + f], alpha * XL[(long long)s * F + f]);
}

// out = elu(out + bias)
__global__ void bias_elu_kernel(float* __restrict__ Out,
                                const float* __restrict__ bias,
                                long long NF, int F) {
  long long idx = (long long)blockIdx.x * blockDim.x + threadIdx.x;
  if (idx >= NF) return;
  float v = Out[idx] + bias[(int)(idx % F)];
  Out[idx] = (v > 0.0f) ? v : expm1f(v);
}

// logits = h @ Wc + bc (8 -> 2), softmax
__global__ void classifier_kernel(const float* __restrict__ Hf,
                                  const float* __restrict__ Wc,
                                  const float* __restrict__ bc,
                                  float* __restrict__ out, int N) {
  int n = blockIdx.x * blockDim.x + threadIdx.x;
  if (n >= N) return;
  float l0 = bc[0], l1 = bc[1];
  const float4* h4 = (const float4*)(Hf + (long long)n * 8);
  float4 h0 = h4[0], h1 = h4[1];
  const float* h = (const float*)&h0;
#pragma unroll
  for (int j = 0; j < 4; ++j) {
    l0 += ((const float*)&h0)[j] * Wc[j * 2 + 0];
    l1 += ((const float*)&h0)[j] * Wc[j * 2 + 1];
  }
#pragma unroll
  for (int j = 0; j < 4; ++j) {
    l0 += ((const float*)&h1)[j] * Wc[(j + 4) * 2 + 0];
    l1 += ((const float*)&h1)[j] * Wc[(j + 4) * 2 + 1];
  }
  (void)h;
  float mx = fmaxf(l0, l1);
  float e0 = expf(l0 - mx), e1 = expf(l1 - mx);
  float inv = 1.0f / (e0 + e1);
  out[(long long)n * 2 + 0] = e0 * inv;
  out[(long long)n * 2 + 1] = e1 * inv;
}

// --------------------------------------------------------------- host side
static inline int cdiv(long long a, long long b) { return (int)((a + b - 1) / b); }

template <int Fin, int H, int C>
static void run_gatv2_layer(const float* Xin,
                            const float* Wl, const float* Wr, const float* We,
                            const float* att, const float* bias,
                            const int* src, const int* dst, const float* ea,
                            const float* loopA,
                            float* XL, float* XR, float* Sbuf, int* Mbuf,
                            float* Dbuf, float* Out,
                            int N, int E, hipStream_t stream) {
  constexpr int F = H * C;
  // node-feature GEMMs (WMMA fp32)
  int tilesM = cdiv(N, 16), tilesN = cdiv(F, 16);
  dim3 blk(32, 4);
  dim3 grd(cdiv((long long)tilesM * tilesN, 4));
  wmma_gemm_kernel<Fin><<<grd, blk, 0, stream>>>(Xin, Wl, XL, N, F, tilesN);
  wmma_gemm_kernel<Fin><<<grd, blk, 0, stream>>>(Xin, Wr, XR, N, F, tilesN);
  // init accumulators
  fill_u32_kernel<<<1024, 256, 0, stream>>>((unsigned*)Mbuf, 0x807FFFFFu, (long long)N * H); // ordf(-inf)
  fill_u32_kernel<<<1024, 256, 0, stream>>>((unsigned*)Dbuf, 0u, (long long)N * H);
  fill_u32_kernel<<<1024, 256, 0, stream>>>((unsigned*)Out,  0u, (long long)N * F);
  // segment softmax + aggregate
  long long totSH = (long long)(E + N) * H;
  edge_score_kernel<H, C><<<cdiv(totSH, 256), 256, 0, stream>>>(
      src, dst, ea, loopA, XL, XR, We, att, Sbuf, Mbuf, E, N);
  edge_exp_kernel<H><<<cdiv(totSH, 256), 256, 0, stream>>>(dst, Sbuf, Mbuf, Dbuf, E, N);
  long long totA = (long long)(E + N) * F;
  edge_aggr_kernel<H, C><<<cdiv(totA, 256), 256, 0, stream>>>(
      src, dst, Sbuf, Dbuf, XL, Out, E, N);
  bias_elu_kernel<<<cdiv((long long)N * F, 256), 256, 0, stream>>>(
      Out, bias, (long long)N * F, F);
}

extern "C" void kernel_launch(void* const* d_in, const int* in_sizes, int n_in,
                              void* d_out, int out_size, void* d_ws, size_t ws_size,
                              hipStream_t stream) {
  const float* x    = (const float*)d_in[0];
  const int*   ei   = (const int*)d_in[1];
  const float* ea   = (const float*)d_in[2];
  const float* Wl1  = (const float*)d_in[3];
  const float* Wr1  = (const float*)d_in[4];
  const float* We1  = (const float*)d_in[5];
  const float* att1 = (const float*)d_in[6];
  const float* b1   = (const float*)d_in[7];
  const float* Wl2  = (const float*)d_in[8];
  const float* Wr2  = (const float*)d_in[9];
  const float* We2  = (const float*)d_in[10];
  const float* att2 = (const float*)d_in[11];
  const float* b2   = (const float*)d_in[12];
  const float* Wl3  = (const float*)d_in[13];
  const float* Wr3  = (const float*)d_in[14];
  const float* We3  = (const float*)d_in[15];
  const float* att3 = (const float*)d_in[16];
  const float* b3   = (const float*)d_in[17];
  const float* Wc   = (const float*)d_in[18];
  const float* bc   = (const float*)d_in[19];

  const int FIN = 128;
  int N = in_sizes[0] / FIN;
  int E = in_sizes[1] / 2;
  const int* src = ei;
  const int* dst = ei + E;

  // workspace layout (floats)
  float* ws = (float*)d_ws;
  float* XL  = ws; ws += (long long)N * 128;
  float* XR  = ws; ws += (long long)N * 128;
  float* F0  = ws; ws += (long long)N * 128;
  float* F1  = ws; ws += (long long)N * 128;
  float* S   = ws; ws += (long long)(E + N) * 4;
  int*   Mb  = (int*)ws; ws += (long long)N * 4;
  float* Db  = ws; ws += (long long)N * 4;
  float* loopA = ws; ws += N;
  float* cntB  = ws; ws += N;
  float* asumB = ws; ws += N;

  // self-loop edge attr = scatter-mean of incoming attrs
  fill_u32_kernel<<<1024, 256, 0, stream>>>((unsigned*)cntB,  0u, (long long)N);
  fill_u32_kernel<<<1024, 256, 0, stream>>>((unsigned*)asumB, 0u, (long long)N);
  sl_accum_kernel<<<cdiv(E, 256), 256, 0, stream>>>(dst, ea, asumB, cntB, E);
  sl_final_kernel<<<cdiv(N, 256), 256, 0, stream>>>(asumB, cntB, loopA, N);

  // layer 1: 128 -> 4 heads x 32 (concat 128)
  run_gatv2_layer<128, 4, 32>(x,  Wl1, Wr1, We1, att1, b1,
                              src, dst, ea, loopA, XL, XR, S, Mb, Db, F0, N, E, stream);
  // layer 2: 128 -> 2 heads x 16 (concat 32)
  run_gatv2_layer<128, 2, 16>(F0, Wl2, Wr2, We2, att2, b2,
                              src, dst, ea, loopA, XL, XR, S, Mb, Db, F1, N, E, stream);
  // layer 3: 32 -> 1 head x 8 (mean over 1 head == identity)
  run_gatv2_layer<32, 1, 8>(F1, Wl3, Wr3, We3, att3, b3,
                            src, dst, ea, loopA, XL, XR, S, Mb, Db, F0, N, E, stream);

  // classifier + softmax -> d_out [N,2] float32
  classifier_kernel<<<cdiv(N, 256), 256, 0, stream>>>(F0, Wc, bc, (float*)d_out, N);
}